// NodeGCN_79053168050214
// MI455X (gfx1250) — compile-verified
//
#include <hip/hip_runtime.h>

// ---------------------------------------------------------------------------
// GCN (2-layer + linear head) for gfx1250.
// GEMMs use V_WMMA_F32_16X16X4_F32 (exact f32, matches reference precision).
// Message passing = edge-parallel gather + global_atomic_add_f32 scatter;
// the 25.6MB feature matrix lives in the 192MB L2, so atomics are L2-local.
// Workspace layout: bufA[N*128] | bufB[N*128] | deg/dis[N]  (~51.5 MB)
// ---------------------------------------------------------------------------

#define FDIM 128

typedef float v2f __attribute__((ext_vector_type(2)));
typedef float v8f __attribute__((ext_vector_type(8)));

__device__ __forceinline__ void atomic_add_f32(float* p, float v) {
  // unsafeAtomicAdd lowers to global_atomic_add_f32 (no CAS loop) on AMD.
  unsafeAtomicAdd(p, v);
}

// ---------------- degree kernels ----------------
__global__ void k_fill(float* __restrict__ p, float val, int n) {
  int i = blockIdx.x * blockDim.x + threadIdx.x;
  if (i < n) p[i] = val;
}

__global__ void k_deg_count(const int* __restrict__ dst, float* __restrict__ deg, int E) {
  int i = blockIdx.x * blockDim.x + threadIdx.x;
  const int stride = gridDim.x * blockDim.x;
  for (; i < E; i += stride) atomic_add_f32(&deg[dst[i]], 1.0f);
}

__global__ void k_rsqrt_inplace(float* __restrict__ deg, int n) {
  int i = blockIdx.x * blockDim.x + threadIdx.x;
  if (i < n) deg[i] = rsqrtf(deg[i]);  // deg >= 1 (self loop), no zero case
}

// ---------------- f32 WMMA GEMM: Out[nrows,128] = (relu?)A[nrows,128] @ W[128,128]
// One wave computes a 16x128 row tile: 8 C-tiles of 16x16, K swept in chunks of 4.
// A layout (ISA 7.12.2, 32-bit A 16x4): lanes 0-15 hold M=lane; v0=K0,v1=K1;
//   lanes 16-31 hold M=lane-16; v0=K2,v1=K3.
// B layout (4x16, rows striped across lanes): v0 = rows {K0 (lanes0-15), K2 (16-31)},
//   v1 = rows {K1, K3}, N = lane&15.
// C/D layout: vgpr v -> row M = v + 8*(lane>=16), col N = lane&15.
__global__ void gcn_gemm_wmma(const float* __restrict__ A,
                              const float* __restrict__ W,
                              float* __restrict__ Out,
                              int nrows, int do_relu) {
  const int wave = blockIdx.x * (blockDim.x >> 5) + (threadIdx.x >> 5);
  const int row0 = wave * 16;
  if (row0 >= nrows) return;
  const int lane = threadIdx.x & 31;
  const int m    = lane & 15;   // row (A) / col (B,C) within tile
  const int kh   = lane >> 4;   // K-half select

  v8f acc[8];
#pragma unroll
  for (int t = 0; t < 8; ++t) { v8f z = {}; acc[t] = z; }

  int ra = row0 + m; if (ra >= nrows) ra = nrows - 1;   // clamp (stores guarded)
  const float* __restrict__ arow = A + (size_t)ra * FDIM;

  for (int kc = 0; kc < FDIM; kc += 4) {
    const int ka = kc + 2 * kh;
    float a0 = arow[ka + 0];
    float a1 = arow[ka + 1];
    if (do_relu) { a0 = fmaxf(a0, 0.0f); a1 = fmaxf(a1, 0.0f); }
    v2f a; a.x = a0; a.y = a1;
#pragma unroll
    for (int nt = 0; nt < 8; ++nt) {
      const int col = nt * 16 + m;
      v2f b;
      b.x = W[(size_t)(ka + 0) * FDIM + col];
      b.y = W[(size_t)(ka + 1) * FDIM + col];
      acc[nt] = __builtin_amdgcn_wmma_f32_16x16x4_f32(
          /*neg_a=*/false, a, /*neg_b=*/false, b,
          /*c_mod=*/(short)0, acc[nt], /*reuse_a=*/false, /*reuse_b=*/false);
    }
  }

#pragma unroll
  for (int nt = 0; nt < 8; ++nt) {
    const int col = nt * 16 + m;
#pragma unroll
    for (int v = 0; v < 8; ++v) {
      const int row = row0 + v + 8 * kh;
      if (row < nrows) Out[(size_t)row * FDIM + col] = acc[nt][v];
    }
  }
}

// ---------------- agg init: bias + self-loop contribution (row-owned, no atomics)
__global__ void k_init_agg(const float* __restrict__ h, const float* __restrict__ dis,
                           const float* __restrict__ bias, float* __restrict__ agg, int n) {
  int i = blockIdx.x * blockDim.x + threadIdx.x;
  const int stride = gridDim.x * blockDim.x;
  const int total = n * FDIM;
  for (; i < total; i += stride) {
    const int node = i >> 7;
    const int f    = i & (FDIM - 1);
    const float d  = dis[node];
    agg[i] = bias[f] + h[i] * d * d;
  }
}

// ---------------- edge scatter: one wave per edge, float4 per lane
__global__ void k_edge_scatter(const int* __restrict__ src, const int* __restrict__ dst,
                               const float* __restrict__ dis, const float* __restrict__ h,
                               float* __restrict__ agg, int E) {
  const int lane = threadIdx.x & 31;
  int w = blockIdx.x * (blockDim.x >> 5) + (threadIdx.x >> 5);
  const int nw = gridDim.x * (blockDim.x >> 5);
  for (int e = w; e < E; e += nw) {
    const int s = src[e];
    const int d = dst[e];
    const float norm = dis[s] * dis[d];
    const float4 hv = ((const float4*)(h + (size_t)s * FDIM))[lane];
    float* ap = agg + (size_t)d * FDIM + lane * 4;
    atomic_add_f32(ap + 0, hv.x * norm);
    atomic_add_f32(ap + 1, hv.y * norm);
    atomic_add_f32(ap + 2, hv.z * norm);
    atomic_add_f32(ap + 3, hv.w * norm);
  }
}

// ---------------- head: out[i] = relu(agg[i]) . Wc + bc  (wave per node)
__global__ void k_classifier(const float* __restrict__ agg, const float* __restrict__ Wc,
                             const float* __restrict__ bc, float* __restrict__ out, int n) {
  const int lane = threadIdx.x & 31;
  int w = blockIdx.x * (blockDim.x >> 5) + (threadIdx.x >> 5);
  const int nw = gridDim.x * (blockDim.x >> 5);
  const float4 wv = ((const float4*)Wc)[lane];
  for (int i = w; i < n; i += nw) {
    const float4 hv = ((const float4*)(agg + (size_t)i * FDIM))[lane];
    float s = fmaxf(hv.x, 0.0f) * wv.x + fmaxf(hv.y, 0.0f) * wv.y +
              fmaxf(hv.z, 0.0f) * wv.z + fmaxf(hv.w, 0.0f) * wv.w;
#pragma unroll
    for (int off = 16; off > 0; off >>= 1) s += __shfl_xor(s, off, 32);
    if (lane == 0) out[i] = s + bc[0];
  }
}

// ---------------------------------------------------------------------------
extern "C" void kernel_launch(void* const* d_in, const int* in_sizes, int n_in,
                              void* d_out, int out_size, void* d_ws, size_t ws_size,
                              hipStream_t stream) {
  const float* x  = (const float*)d_in[0];
  const int*   ei = (const int*)d_in[1];
  const float* W1 = (const float*)d_in[2];
  const float* b1 = (const float*)d_in[3];
  const float* W2 = (const float*)d_in[4];
  const float* b2 = (const float*)d_in[5];
  const float* Wc = (const float*)d_in[6];
  const float* bc = (const float*)d_in[7];
  float* out = (float*)d_out;

  const int N = in_sizes[0] / FDIM;
  const int E = in_sizes[1] / 2;      // edge_index is [2, E] flat: src then dst
  const int* src = ei;
  const int* dst = ei + E;

  float* bufA = (float*)d_ws;                   // [N*128] GEMM output h
  float* bufB = bufA + (size_t)N * FDIM;        // [N*128] aggregation buffer
  float* dis  = bufB + (size_t)N * FDIM;        // [N] deg -> deg^-1/2 in place

  // ---- symmetric normalization ----
  k_fill<<<(N + 255) / 256, 256, 0, stream>>>(dis, 1.0f, N);       // self loop
  k_deg_count<<<2048, 256, 0, stream>>>(dst, dis, E);
  k_rsqrt_inplace<<<(N + 255) / 256, 256, 0, stream>>>(dis, N);

  const int ntiles = (N + 15) / 16;
  const int gemm_blocks = (ntiles + 7) / 8;     // 8 waves / 256-thread block

  // ---- layer 1 ----
  gcn_gemm_wmma<<<gemm_blocks, 256, 0, stream>>>(x, W1, bufA, N, /*relu=*/0);
  k_init_agg<<<2048, 256, 0, stream>>>(bufA, dis, b1, bufB, N);
  k_edge_scatter<<<8192, 256, 0, stream>>>(src, dst, dis, bufA, bufB, E);

  // ---- layer 2 (ReLU of layer-1 agg fused into GEMM A load) ----
  gcn_gemm_wmma<<<gemm_blocks, 256, 0, stream>>>(bufB, W2, bufA, N, /*relu=*/1);
  k_init_agg<<<2048, 256, 0, stream>>>(bufA, dis, b2, bufB, N);
  k_edge_scatter<<<8192, 256, 0, stream>>>(src, dst, dis, bufA, bufB, E);

  // ---- linear head (ReLU fused) ----
  k_classifier<<<4096, 256, 0, stream>>>(bufB, Wc, bc, out, N);
}